// GCN_36240934043968
// MI455X (gfx1250) — compile-verified
//
#include <hip/hip_runtime.h>
#include <hip/hip_bf16.h>

typedef float v2f __attribute__((ext_vector_type(2)));
typedef float v8f __attribute__((ext_vector_type(8)));
typedef unsigned int u32x4 __attribute__((ext_vector_type(4)));
typedef int i32x4 __attribute__((ext_vector_type(4)));
typedef int i32x8 __attribute__((ext_vector_type(8)));

#define GDIM 256
#define RREL 8
#define BBAS 30
#define KC 64
#define A_LDS 68   // 64 + 4 pad floats: TDM pad_interval=64DW, pad_amount=4DW
#define B_LDS 68

// ---------------------------------------------------------------------------
// Tensor Data Mover: DMA a 2D tile (tileW x tileH f32, row stride strideElems)
// from global memory into LDS at ldsOff, inserting 4 dwords of padding after
// every 64 dwords (=> LDS row stride 68 floats, bank-conflict-free frags).
// Descriptor packing per CDNA5 ISA ch. 8 (D# groups 0/1); groups 2/3 zero.
__device__ __forceinline__ void tdm_load_tile(const float* gptr, unsigned ldsOff,
                                              unsigned tileW, unsigned tileH,
                                              unsigned strideElems, unsigned tdim0) {
    const unsigned long long ga = (unsigned long long)(uintptr_t)gptr;
    u32x4 g0;
    g0[0] = 1u;                                        // count=1 (valid), user mode
    g0[1] = ldsOff;                                    // lds_addr (bytes)
    g0[2] = (unsigned)(ga & 0xffffffffu);              // global_addr[31:0]
    g0[3] = (unsigned)((ga >> 32) & 0x1ffffffu)        // global_addr[56:32]
          | (2u << 30);                                // type = 2 ("image")
    const unsigned td1 = 1u << 20;                     // oversized: tiles in-bounds
    i32x8 g1;
    g1[0] = (int)((2u << 16)                           // data_size = 4 bytes
                | (1u << 20)                           // pad_enable
                | (5u << 22)                           // pad_interval: 64 dwords
                | (3u << 25));                         // pad_amount: 4 dwords
    g1[1] = (int)((tdim0 & 0xffffu) << 16);            // tensor_dim0[15:0]
    g1[2] = (int)((tdim0 >> 16) | ((td1 & 0xffffu) << 16));
    g1[3] = (int)((td1 >> 16) | (tileW << 16));        // tile_dim0
    g1[4] = (int)(tileH & 0xffffu);                    // tile_dim1 (tile_dim2 = 0)
    g1[5] = (int)strideElems;                          // tensor_dim0_stride[31:0]
    g1[6] = 0;
    g1[7] = 0;
    const i32x4 z4 = {0, 0, 0, 0};
#if defined(__clang_major__) && (__clang_major__ >= 23)
    const i32x8 z8 = {0, 0, 0, 0, 0, 0, 0, 0};
    __builtin_amdgcn_tensor_load_to_lds(g0, g1, z4, z4, z8, 0);
#else
    __builtin_amdgcn_tensor_load_to_lds(g0, g1, z4, z4, 0);
#endif
}

// ---------------------------------------------------------------------------
// Zero a float range (also used for int ranges). n must be a multiple of 4.
__global__ void zero_f32_k(float* __restrict__ p, long long n) {
    long long i = ((long long)blockIdx.x * blockDim.x + threadIdx.x) * 4;
    const long long stride = (long long)gridDim.x * blockDim.x * 4;
    for (; i < n; i += stride) {
        *(float4*)(p + i) = make_float4(0.f, 0.f, 0.f, 0.f);
    }
}

// ---------------------------------------------------------------------------
// W_full[r, i, o] = sum_b comp[r, b] * basis[b, i, o]      (R x G x H1)
__global__ void wfull_k(const float* __restrict__ comp,
                        const float* __restrict__ basis,
                        float* __restrict__ wfull) {
    const int idx = blockIdx.x * blockDim.x + threadIdx.x;   // r*65536 + io
    const int r  = idx >> 16;
    const int io = idx & 65535;
    float acc = 0.f;
#pragma unroll 6
    for (int b = 0; b < BBAS; ++b)
        acc += comp[r * BBAS + b] * basis[b * 65536 + io];
    wfull[idx] = acc;
}

// ---------------------------------------------------------------------------
// cnt[tgt*R + rel] += 1 per edge (PyG 'mean' per-relation aggregation).
__global__ void count_k(const int* __restrict__ edge_index,
                        const int* __restrict__ edge_type,
                        int* __restrict__ cnt, int E) {
    const int e = blockIdx.x * blockDim.x + threadIdx.x;
    if (e >= E) return;
    const int t = edge_index[E + e];
    const int r = edge_type[e];
    atomicAdd(&cnt[t * RREL + r], 1);
}

// ---------------------------------------------------------------------------
// Edge scatter. mode 0: Y[(tgt*R+rel), :] += x[src, :] / cnt[tgt,rel]
//               mode 1: Out[tgt, :]      += edge_norm[e] * X[src, :]
__global__ void scatter_k(const float* __restrict__ X,
                          const int* __restrict__ edge_index,
                          const int* __restrict__ edge_type,
                          const int* __restrict__ cnt,
                          const float* __restrict__ edge_norm,
                          float* __restrict__ Out, int E, int mode) {
    const long long gid = (long long)blockIdx.x * blockDim.x + threadIdx.x;
    const int e = (int)(gid >> 6);
    const int c = ((int)gid & 63) * 4;
    if (e >= E) return;
    const int s = edge_index[e];
    const int t = edge_index[E + e];
    float scale;
    long long row;
    if (mode == 0) {
        const int r = edge_type[e];
        row = (long long)(t * RREL + r);
        scale = 1.0f / (float)cnt[t * RREL + r];
    } else {
        row = t;
        scale = edge_norm[e];
    }
    const float4 v = *(const float4*)(X + (long long)s * GDIM + c);
    float* dst = Out + row * GDIM + c;
    atomicAdd(dst + 0, v.x * scale);
    atomicAdd(dst + 1, v.y * scale);
    atomicAdd(dst + 2, v.z * scale);
    atomicAdd(dst + 3, v.w * scale);
}

// ---------------------------------------------------------------------------
// Two-source WMMA GEMM, Ncols fixed at 256, TDM-staged, double-buffered LDS.
//   acc = A1[M,K1] @ B1[K1,256] + A2[M,K2] @ B2[K2,256]
//   mode 0: Out = acc + bias
//   mode 1: s = sigmoid(acc + bias); Out = s*Hnew + (1-s)*Hold
// Block = 256 threads = 8 waves; block tile 32(M) x 64(N); wave tile 16x16.
__global__ __launch_bounds__(256) void gemm_wmma_k(
    const float* __restrict__ A1, int K1, int lda1, const float* __restrict__ B1,
    const float* __restrict__ A2, int K2, int lda2, const float* __restrict__ B2,
    const float* __restrict__ bias,
    const float* __restrict__ Hnew, const float* __restrict__ Hold,
    float* __restrict__ Out, int mode) {
    __shared__ float As[2][32 * A_LDS];
    __shared__ float Bs[2][KC * B_LDS];

    const int tid  = threadIdx.x;
    const int lane = tid & 31;         // wave32
    const int wave = tid >> 5;
    const int wm = wave & 1;           // M subtile within block (0..1)
    const int wn = wave >> 1;          // N subtile within block (0..3)
    const int blockM = blockIdx.x;     // 32 rows per block
    const int blockN = blockIdx.y;     // 64 cols per block

    v8f acc = {};

    const int mrow = wm * 16 + (lane & 15);  // A frag: lane -> M row
    const int half = lane >> 4;              // A/B frag: lane half -> K offset
    const int ncol = wn * 16 + (lane & 15);  // B frag: lane -> N col

    const int nCh1 = K1 / KC;
    const int nCh2 = (A2 != nullptr) ? (K2 / KC) : 0;
    const int totalCh = nCh1 + nCh2;

    // Issue both TDM tile loads (A then B) for virtual chunk ci into buffer buf.
    auto issue = [&](int ci, int buf) {
        const float* A;
        const float* B;
        int lda, kb;
        if (ci < nCh1) { A = A1; B = B1; lda = lda1; kb = ci * KC; }
        else           { A = A2; B = B2; lda = lda2; kb = (ci - nCh1) * KC; }
        tdm_load_tile(A + (long long)(blockM * 32) * lda + kb,
                      (unsigned)(uintptr_t)&As[buf][0], KC, 32,
                      (unsigned)lda, (unsigned)lda);
        tdm_load_tile(B + (long long)kb * 256 + blockN * 64,
                      (unsigned)(uintptr_t)&Bs[buf][0], KC, KC, 256u, 256u);
    };

    if (wave == 0) issue(0, 0);

    for (int ci = 0; ci < totalCh; ++ci) {
        const int cur = ci & 1;
        if (wave == 0) {
            if (ci + 1 < totalCh) {
                issue(ci + 1, cur ^ 1);
                __builtin_amdgcn_s_wait_tensorcnt(2);   // retire chunk ci
            } else {
                __builtin_amdgcn_s_wait_tensorcnt(0);
            }
        }
        __syncthreads();

        const float* As_c = &As[cur][0];
        const float* Bs_c = &Bs[cur][0];
        // Preload all fragments for the chunk, then run the WMMA chain
        // back-to-back (single dscnt wait instead of one per WMMA).
        v2f af[16], bf[16];
#pragma unroll
        for (int t = 0; t < 16; ++t) {
            const int k0 = t * 4 + half * 2;
            af[t] = *(const v2f*)(As_c + mrow * A_LDS + k0);
            bf[t].x = Bs_c[k0 * B_LDS + ncol];
            bf[t].y = Bs_c[(k0 + 1) * B_LDS + ncol];
        }
#pragma unroll
        for (int t = 0; t < 16; ++t) {
            acc = __builtin_amdgcn_wmma_f32_16x16x4_f32(
                false, af[t], false, bf[t], (short)0, acc, false, false);
        }
        __syncthreads();
    }

    // Epilogue. C/D layout: lane n = lane&15; VGPR j -> M = 8*(lane>>4) + j.
    const int nloc  = lane & 15;
    const int mbase = (lane >> 4) * 8;
    const int gn = blockN * 64 + wn * 16 + nloc;
    const float bn = bias ? bias[gn] : 0.0f;
#pragma unroll
    for (int j = 0; j < 8; ++j) {
        const long long gm  = (long long)blockM * 32 + wm * 16 + mbase + j;
        const long long idx = gm * 256 + gn;
        float v = acc[j] + bn;
        if (mode == 1) {
            const float s = 1.0f / (1.0f + __expf(-v));
            v = s * Hnew[idx] + (1.0f - s) * Hold[idx];
        }
        Out[idx] = v;
    }
}

// ---------------------------------------------------------------------------
extern "C" void kernel_launch(void* const* d_in, const int* in_sizes, int n_in,
                              void* d_out, int out_size, void* d_ws, size_t ws_size,
                              hipStream_t stream) {
    const int N = 20000, E = 320000;

    const float* x        = (const float*)d_in[0];
    const int*   ei       = (const int*)d_in[1];
    const float* enorm    = (const float*)d_in[2];
    const int*   etype    = (const int*)d_in[3];
    const float* basis    = (const float*)d_in[4];
    const float* comp     = (const float*)d_in[5];
    const float* root1    = (const float*)d_in[6];
    const float* bias1    = (const float*)d_in[7];
    const float* w_rel    = (const float*)d_in[8];
    const float* b_rel    = (const float*)d_in[9];
    const float* w_root   = (const float*)d_in[10];
    const float* gate1_w  = (const float*)d_in[11];
    const float* gate1_b  = (const float*)d_in[12];
    const float* gate2_w  = (const float*)d_in[13];
    const float* gate2_b  = (const float*)d_in[14];
    float* out = (float*)d_out;

    // Workspace layout (float units)
    float* ws = (float*)d_ws;
    const long long WFULL_SZ = (long long)RREL * GDIM * 256;   //   524288
    const long long CNT_SZ   = (long long)N * RREL;            //   160000
    const long long Y_SZ     = (long long)N * RREL * GDIM;     // 40960000
    const long long H_SZ     = (long long)N * 256;             //  5120000
    float* wfull  = ws;
    int*   cnt    = (int*)(ws + WFULL_SZ);
    float* Y      = ws + WFULL_SZ + CNT_SZ;
    float* h1_gcn = Y + Y_SZ;
    float* h1     = h1_gcn + H_SZ;
    float* agg2   = h1 + H_SZ;
    float* h2_gcn = agg2 + H_SZ;

    const dim3 gGrid(625, 4);   // 20000/32 M blocks x 256/64 N blocks
    const int  TB = 256;

    // 1. Zero cnt + Y (contiguous) and agg2
    zero_f32_k<<<2048, TB, 0, stream>>>((float*)cnt, CNT_SZ + Y_SZ);
    zero_f32_k<<<1024, TB, 0, stream>>>(agg2, H_SZ);

    // 2. W_full = comp @ basis
    wfull_k<<<(int)(WFULL_SZ / TB), TB, 0, stream>>>(comp, basis, wfull);

    // 3. Per-(tgt, rel) edge counts for mean aggregation
    count_k<<<(E + TB - 1) / TB, TB, 0, stream>>>(ei, etype, cnt, E);

    // 4. Aggregate-then-transform: Y[tgt*R+rel] += x[src]/cnt
    scatter_k<<<(E * 64) / TB, TB, 0, stream>>>(x, ei, etype, cnt, enorm, Y, E, 0);

    // 5. h1_gcn = Y @ W_full(2048x256) + x @ root1 + bias1
    gemm_wmma_k<<<gGrid, TB, 0, stream>>>(Y, RREL * GDIM, RREL * GDIM, wfull,
                                          x, GDIM, GDIM, root1,
                                          bias1, nullptr, nullptr, h1_gcn, 0);

    // 6. Gate 1: h1 = sigmoid([x|h1_gcn]@gate1_w + b) * h1_gcn + (1-.)*x
    gemm_wmma_k<<<gGrid, TB, 0, stream>>>(x, GDIM, GDIM, gate1_w,
                                          h1_gcn, 256, 256, gate1_w + 256 * 256,
                                          gate1_b, h1_gcn, x, h1, 1);

    // 7. agg2[tgt] += edge_norm * h1[src]
    scatter_k<<<(E * 64) / TB, TB, 0, stream>>>(h1, ei, etype, cnt, enorm, agg2, E, 1);

    // 8. h2_gcn = agg2 @ w_rel + h1 @ w_root + b_rel
    gemm_wmma_k<<<gGrid, TB, 0, stream>>>(agg2, 256, 256, w_rel,
                                          h1, 256, 256, w_root,
                                          b_rel, nullptr, nullptr, h2_gcn, 0);

    // 9. Gate 2 -> final output
    gemm_wmma_k<<<gGrid, TB, 0, stream>>>(h1, 256, 256, gate2_w,
                                          h2_gcn, 256, 256, gate2_w + 256 * 256,
                                          gate2_b, h2_gcn, h1, out, 1);
}